// ClasswiseECELossSeg_2396591751307
// MI455X (gfx1250) — compile-verified
//
#include <hip/hip_runtime.h>
#include <stdint.h>

// Problem constants (from reference): logits [4,19,512,1024] f32, labels [4,512,1024] i32
#define NCLS   19
#define NBINS  15
#define HW     (512 * 1024)          // 524288 = 1<<19
#define NPIX   (4 * HW)              // 2097152 pixels
#define TILE   512                   // pixels staged per block-iteration (2 per thread)
#define BLOCK  256                   // 8 wave32s
#define NWAVES (BLOCK / 32)
#define HISTSZ (NCLS * NBINS)        // 285
#define NTILES (NPIX / TILE)         // 4096
#define HIST_BLOCKS 1024

// Native CDNA5 no-return LDS float atomic (avoids any CAS-loop lowering).
__device__ __forceinline__ void lds_fadd(float* p, float v) {
    unsigned a = (unsigned)(uintptr_t)p;          // low 32 bits = LDS byte offset
    asm volatile("ds_add_f32 %0, %1" :: "v"(a), "v"(v) : "memory");
}

__global__ __launch_bounds__(BLOCK) void ece_zero(float* __restrict__ g) {
    int i = blockIdx.x * blockDim.x + threadIdx.x;
    if (i < HISTSZ) g[i] = 0.0f;
}

__global__ __launch_bounds__(BLOCK) void ece_hist(const float* __restrict__ logits,
                                                  const int*   __restrict__ labels,
                                                  float*       __restrict__ gHist) {
    // Double-buffered async staging: [buf][channel][pixel-in-tile]
    __shared__ float stage[2][NCLS][TILE];           // 77824 B
    __shared__ float whist[NWAVES][HISTSZ];          // 9120 B, per-wave copies

    const int tid  = threadIdx.x;
    const int wave = tid >> 5;
    const int lane = tid & 31;

    for (int i = tid; i < NWAVES * HISTSZ; i += BLOCK) (&whist[0][0])[i] = 0.0f;
    __syncthreads();

    // Per-lane register accumulators for the dominant bin-0 contributions:
    // reduced across lanes only once, at kernel end (no per-tile shuffles).
    float acc0[NCLS];
#pragma unroll
    for (int c = 0; c < NCLS; ++c) acc0[c] = 0.0f;

    const int stride = gridDim.x;
    int t = blockIdx.x;

    // Issue 19 per-lane async B64 copies (one per channel) for one tile.
    // Each wave instruction moves 256 contiguous bytes from a channel plane.
    auto issue = [&](int tile, int buf) {
        int p = tile * TILE + 2 * tid;   // 2 consecutive pixels per thread
        int b = p >> 19;                 // batch index (HW = 1<<19)
        int q = p & (HW - 1);            // position within plane (same for both pixels)
        const float* gp0 = logits + ((size_t)(b * NCLS) << 19) + q;
        unsigned l0 = (unsigned)(uintptr_t)&stage[buf][0][2 * tid];  // LDS byte offset
#pragma unroll
        for (int c = 0; c < NCLS; ++c) {
            uint64_t ga = (uint64_t)(uintptr_t)(gp0 + ((size_t)c << 19));
            unsigned la = l0 + (unsigned)(c * (TILE * 4));
            asm volatile("global_load_async_to_lds_b64 %0, %1, off"
                         :: "v"(la), "v"(ga) : "memory");            // ASYNCcnt += 1
        }
    };

    if (t < NTILES) issue(t, 0);
    int it = 0;
    for (; t < NTILES; t += stride, ++it) {
        const int cur = it & 1;
        const int nt  = t + stride;

        const int p0 = t * TILE + 2 * tid;
        const int2 lab2 = *(const int2*)&labels[p0];              // issue before the wait

        if (nt < NTILES) {
            issue(nt, cur ^ 1);                                   // 38 in flight
            asm volatile("s_wait_asynccnt 19" ::: "memory");      // current tile done
        } else {
            asm volatile("s_wait_asynccnt 0" ::: "memory");
        }

        // Single-pass softmax for both pixels; all values live in registers.
        float2 v[NCLS];
        float mA = -3.402823466e38f, mB = -3.402823466e38f;
#pragma unroll
        for (int c = 0; c < NCLS; ++c) {
            float2 e = *(const float2*)&stage[cur][c][2 * tid];   // ds_load_b64
            v[c] = e;
            mA = fmaxf(mA, e.x);
            mB = fmaxf(mB, e.y);
        }
        float sA = 0.0f, sB = 0.0f;
#pragma unroll
        for (int c = 0; c < NCLS; ++c) {
            float eA = __expf(v[c].x - mA);
            float eB = __expf(v[c].y - mB);
            v[c].x = eA; v[c].y = eB;
            sA += eA; sB += eB;
        }
        const float iA = 1.0f / sA, iB = 1.0f / sB;

#pragma unroll
        for (int c = 0; c < NCLS; ++c) {
            float confA = v[c].x * iA;
            float confB = v[c].y * iB;

            int binA = (int)ceilf(confA * (float)NBINS) - 1;      // bin i = (i/15,(i+1)/15]
            int binB = (int)ceilf(confB * (float)NBINS) - 1;
            binA = binA < 0 ? 0 : (binA > NBINS - 1 ? NBINS - 1 : binA);
            binB = binB < 0 ? 0 : (binB > NBINS - 1 ? NBINS - 1 : binB);

            const bool validA = confA > 0.0f;                     // conf==0 -> dumped
            const bool validB = confB > 0.0f;
            const float valA = validA ? (confA - ((lab2.x == c) ? 1.0f : 0.0f)) : 0.0f;
            const float valB = validB ? (confB - ((lab2.y == c) ? 1.0f : 0.0f)) : 0.0f;

            const bool hotA = validA && (binA != 0);
            const bool hotB = validB && (binB != 0);

            // Dominant bin-0 (and invalid->0) contributions stay in registers.
            acc0[c] += (hotA ? 0.0f : valA) + (hotB ? 0.0f : valB);

            // ~25% of lanes, spread over 14 addresses: native no-return ds_add_f32.
            if (hotA) lds_fadd(&whist[wave][c * NBINS + binA], valA);
            if (hotB) lds_fadd(&whist[wave][c * NBINS + binB], valB);
        }
    }

    // One-time cross-lane reduction of the register accumulators.
#pragma unroll
    for (int c = 0; c < NCLS; ++c) {
        float r = acc0[c];
        r += __shfl_xor(r, 1);
        r += __shfl_xor(r, 2);
        r += __shfl_xor(r, 4);
        r += __shfl_xor(r, 8);
        r += __shfl_xor(r, 16);
        if (lane == 0) lds_fadd(&whist[wave][c * NBINS], r);
    }

    __syncthreads();
    for (int i = tid; i < HISTSZ; i += BLOCK) {
        float acc = 0.0f;
#pragma unroll
        for (int w = 0; w < NWAVES; ++w) acc += whist[w][i];
        unsafeAtomicAdd(&gHist[i], acc);                          // global_atomic_add_f32
    }
}

__global__ __launch_bounds__(32) void ece_final(const float* __restrict__ gHist,
                                                float* __restrict__ out) {
    const int lane = threadIdx.x;
    float s = 0.0f;
    for (int i = lane; i < HISTSZ; i += 32) s += fabsf(gHist[i]);
    s += __shfl_xor(s, 1);
    s += __shfl_xor(s, 2);
    s += __shfl_xor(s, 4);
    s += __shfl_xor(s, 8);
    s += __shfl_xor(s, 16);
    if (lane == 0)
        out[0] = s * (1.0f / ((float)NPIX * (float)NCLS));  // sce = sum|D| / (N*C)
}

extern "C" void kernel_launch(void* const* d_in, const int* in_sizes, int n_in,
                              void* d_out, int out_size, void* d_ws, size_t ws_size,
                              hipStream_t stream) {
    const float* logits = (const float*)d_in[0];
    const int*   labels = (const int*)d_in[1];
    float* gHist = (float*)d_ws;     // 285 floats of scratch

    ece_zero<<<(HISTSZ + BLOCK - 1) / BLOCK, BLOCK, 0, stream>>>(gHist);
    ece_hist<<<HIST_BLOCKS, BLOCK, 0, stream>>>(logits, labels, gHist);
    ece_final<<<1, 32, 0, stream>>>(gHist, (float*)d_out);
}